// Attention_44289702756433
// MI455X (gfx1250) — compile-verified
//
#include <hip/hip_runtime.h>
#include <float.h>

// ---------------------------------------------------------------------------
// Shapes (fixed by the reference)
// ---------------------------------------------------------------------------
#define BATCH   32
#define SEQ     1024
#define H2DIM   1024
#define TOPICD  256
#define KTOT    (H2DIM + TOPICD)   // 1280: fused [s | hidden_z] @ [W_s ; W_z]

typedef __attribute__((ext_vector_type(2))) float v2f;
typedef __attribute__((ext_vector_type(8))) float v8f;

// ---------------------------------------------------------------------------
// Fast device math (gfx1250: v_exp_f32 / v_rcp_f32 are single trans ops)
// ---------------------------------------------------------------------------
__device__ __forceinline__ float fast_tanh(float x) {
    // tanh(x) = (e^{2x}-1)/(e^{2x}+1); clamp so exp2 stays in range.
    float xc = fminf(fmaxf(x, -15.0f), 15.0f);
    float t  = __builtin_amdgcn_exp2f(xc * 2.8853900817779268f); // 2*log2(e)
    return (t - 1.0f) * __builtin_amdgcn_rcpf(t + 1.0f);
}

__device__ __forceinline__ float wave_reduce_add(float x) {
    #pragma unroll
    for (int off = 16; off > 0; off >>= 1) x += __shfl_xor(x, off, 32);
    return x;
}
__device__ __forceinline__ float wave_reduce_max(float x) {
    #pragma unroll
    for (int off = 16; off > 0; off >>= 1) x = fmaxf(x, __shfl_xor(x, off, 32));
    return x;
}

// ---------------------------------------------------------------------------
// Kernel 1: pre[b,h] = (s @ W_s)[b,h] + (hidden_z @ W_z)[b,h]
// Fused GEMM [32 x 1280] x [1280 x 1024] on V_WMMA_F32_16X16X4_F32.
// One wave per 16x16 output tile: 2 M-tiles x 64 N-tiles = 128 waves.
// A 16x4 f32 layout (ISA 7.12.2): lane%16 = M row; lane<16 holds K={0,1},
// lane>=16 holds K={2,3} in VGPRs {0,1}. B 4x16 mirrors with N in place of M.
// C/D: VGPR r -> M = r (lanes 0-15) / r+8 (lanes 16-31), N = lane%16.
// ---------------------------------------------------------------------------
__global__ void __launch_bounds__(128)
proj_wmma_kernel(const float* __restrict__ s,    // [32,1024]
                 const float* __restrict__ hz,   // [32,256]
                 const float* __restrict__ Ws,   // [1024,1024] (in,out)
                 const float* __restrict__ Wz,   // [256,1024]  (in,out)
                 float* __restrict__ pre)        // [32,1024]
{
    const int waveId = (blockIdx.x * blockDim.x + threadIdx.x) >> 5;
    const int lane   = threadIdx.x & 31;
    const int mtile  = waveId & 1;        // 32 rows -> 2 tiles
    const int ntile  = waveId >> 1;       // 1024 cols -> 64 tiles
    const int lm     = lane & 15;
    const int kh     = (lane >> 4) << 1;  // 0 or 2: K-pair selector
    const int mrow   = mtile * 16 + lm;
    const int ncol   = ntile * 16 + lm;

    v8f acc = {};

    // K phase 1: s @ W_s  (K = 0..1023)
    for (int k0 = 0; k0 < H2DIM; k0 += 4) {
        const int k = k0 + kh;
        v2f a; a.x = s[mrow * H2DIM + k];
               a.y = s[mrow * H2DIM + k + 1];
        v2f w; w.x = Ws[(size_t)k       * H2DIM + ncol];
               w.y = Ws[(size_t)(k + 1) * H2DIM + ncol];
        acc = __builtin_amdgcn_wmma_f32_16x16x4_f32(
            false, a, false, w, (short)0, acc, false, false);
    }
    // K phase 2: hidden_z @ W_z  (K = 1024..1279)
    for (int k0 = 0; k0 < TOPICD; k0 += 4) {
        const int k = k0 + kh;
        v2f a; a.x = hz[mrow * TOPICD + k];
               a.y = hz[mrow * TOPICD + k + 1];
        v2f w; w.x = Wz[(size_t)k       * H2DIM + ncol];
               w.y = Wz[(size_t)(k + 1) * H2DIM + ncol];
        acc = __builtin_amdgcn_wmma_f32_16x16x4_f32(
            false, a, false, w, (short)0, acc, false, false);
    }

    const int rbase = mtile * 16 + ((lane >> 4) ? 8 : 0);
    #pragma unroll
    for (int r = 0; r < 8; ++r) {
        pre[(rbase + r) * H2DIM + ncol] = acc[r];
    }
}

// ---------------------------------------------------------------------------
// Kernel 2: e_t[b,s] = sum_h tanh(enc[b,s,h] + pre[b,h] + cov[b,s]*Wc[h] + bc[h]) * v[h]
// One wave per (b,s) row. 128 MB streamed via coalesced float4 loads — this is
// the bandwidth-bound hot loop (~5.5 us floor at 23.3 TB/s).
// ---------------------------------------------------------------------------
__global__ void __launch_bounds__(256)
score_kernel(const float* __restrict__ enc,  // [32,1024,1024]
             const float* __restrict__ pre,  // [32,1024]
             const float* __restrict__ cov,  // [32,1024]
             const float* __restrict__ Wc,   // [1024]
             const float* __restrict__ bc,   // [1024]
             const float* __restrict__ v,    // [1024]
             float* __restrict__ e_t)        // [32,1024]
{
    const int wave = (blockIdx.x * blockDim.x + threadIdx.x) >> 5;  // row id
    const int lane = threadIdx.x & 31;
    const int b    = wave >> 10;          // / SEQ

    const float4* er = (const float4*)(enc + (size_t)wave * H2DIM);
    const float4* pr = (const float4*)(pre + (size_t)b * H2DIM);
    const float4* wc = (const float4*)Wc;
    const float4* bb = (const float4*)bc;
    const float4* vv = (const float4*)v;
    const float   c  = cov[wave];

    float acc = 0.0f;
    #pragma unroll
    for (int i = 0; i < 8; ++i) {                 // 8 * 32 lanes * 4 = 1024
        const int h4 = i * 32 + lane;             // coalesced 512B per wave
        float4 e = er[h4];
        float4 p = pr[h4];
        float4 w = wc[h4];
        float4 o = bb[h4];
        float4 u = vv[h4];
        acc += fast_tanh(e.x + p.x + fmaf(c, w.x, o.x)) * u.x;
        acc += fast_tanh(e.y + p.y + fmaf(c, w.y, o.y)) * u.y;
        acc += fast_tanh(e.z + p.z + fmaf(c, w.z, o.z)) * u.z;
        acc += fast_tanh(e.w + p.w + fmaf(c, w.w, o.w)) * u.w;
    }
    acc = wave_reduce_add(acc);
    if (lane == 0) e_t[wave] = acc;
}

// ---------------------------------------------------------------------------
// Kernel 3: masked softmax over S per batch + coverage update.
// 32 blocks x 256 threads (8 wave32s); 4 elements/thread.
// ---------------------------------------------------------------------------
__global__ void __launch_bounds__(256)
softmax_kernel(const float* __restrict__ e_t,   // [32,1024]
               const int*   __restrict__ mask,  // [32,1024]
               const float* __restrict__ cov,   // [32,1024]
               float* __restrict__ a_out,       // [32,1024]
               float* __restrict__ nc_out)      // [32,1024]
{
    __shared__ float sred[8];
    const int b    = blockIdx.x;
    const int tid  = threadIdx.x;
    const int wid  = tid >> 5;
    const int lane = tid & 31;

    const float* e = e_t  + (size_t)b * SEQ;
    const int*   m = mask + (size_t)b * SEQ;
    const float* c = cov  + (size_t)b * SEQ;

    float vals[4];
    float lmax = -FLT_MAX;
    #pragma unroll
    for (int i = 0; i < 4; ++i) {
        const int idx = tid + i * 256;
        const float x = (m[idx] == 0) ? -FLT_MAX : e[idx];
        vals[i] = x;
        lmax = fmaxf(lmax, x);
    }

    // block max
    float r = wave_reduce_max(lmax);
    if (lane == 0) sred[wid] = r;
    __syncthreads();
    if (wid == 0) {
        float y = (lane < 8) ? sred[lane] : -FLT_MAX;
        y = wave_reduce_max(y);
        if (lane == 0) sred[0] = y;
    }
    __syncthreads();
    const float bmax = sred[0];
    __syncthreads();

    // exp + block sum (branchless: (-FLT_MAX - bmax) underflows exp2 to 0)
    float lsum = 0.0f;
    #pragma unroll
    for (int i = 0; i < 4; ++i) {
        const float ex = __builtin_amdgcn_exp2f((vals[i] - bmax) * 1.4426950408889634f);
        vals[i] = ex;
        lsum += ex;
    }
    r = wave_reduce_add(lsum);
    if (lane == 0) sred[wid] = r;
    __syncthreads();
    if (wid == 0) {
        float y = (lane < 8) ? sred[lane] : 0.0f;
        y = wave_reduce_add(y);
        if (lane == 0) sred[0] = y;
    }
    __syncthreads();
    const float inv = 1.0f / sred[0];

    #pragma unroll
    for (int i = 0; i < 4; ++i) {
        const int idx = tid + i * 256;
        const float a = vals[i] * inv;
        a_out[(size_t)b * SEQ + idx]  = a;
        nc_out[(size_t)b * SEQ + idx] = c[idx] + a;
    }
}

// ---------------------------------------------------------------------------
// Launcher
// Inputs (setup_inputs order):
//  0 encoder_features [32,1024,1024] f32
//  1 mask             [32,1024]      i32
//  2 s                [32,1024]      f32
//  3 coverage         [32,1024]      f32
//  4 hidden_z         [32,256]       f32
//  5 W_s              [1024,1024]    f32
//  6 W_c              [1,1024]       f32
//  7 b_c              [1024]         f32
//  8 W_z              [256,1024]     f32
//  9 v                [1024]         f32
// Output: a_t [32,1024] then next_coverage [32,1024], f32.
// ---------------------------------------------------------------------------
extern "C" void kernel_launch(void* const* d_in, const int* in_sizes, int n_in,
                              void* d_out, int out_size, void* d_ws, size_t ws_size,
                              hipStream_t stream) {
    const float* enc = (const float*)d_in[0];
    const int*   msk = (const int*)  d_in[1];
    const float* s   = (const float*)d_in[2];
    const float* cov = (const float*)d_in[3];
    const float* hz  = (const float*)d_in[4];
    const float* Ws  = (const float*)d_in[5];
    const float* Wc  = (const float*)d_in[6];
    const float* bc  = (const float*)d_in[7];
    const float* Wz  = (const float*)d_in[8];
    const float* v   = (const float*)d_in[9];

    float* a_out  = (float*)d_out;                  // [32,1024]
    float* nc_out = a_out + BATCH * SEQ;            // [32,1024]

    float* pre = (float*)d_ws;                      // [32,1024]
    float* e_t = pre + BATCH * H2DIM;               // [32,1024]

    // 1) fused projection on WMMA: 128 waves = 32 blocks x 4 waves
    proj_wmma_kernel<<<32, 128, 0, stream>>>(s, hz, Ws, Wz, pre);

    // 2) bandwidth-bound scoring: one wave per row, 32768 waves
    {
        const int waves_per_block = 8;                         // 256 threads
        const int rows = BATCH * SEQ;                          // 32768
        score_kernel<<<rows / waves_per_block, 256, 0, stream>>>(
            enc, pre, cov, Wc, bc, v, e_t);
    }

    // 3) softmax + coverage update: one block per batch row
    softmax_kernel<<<BATCH, 256, 0, stream>>>(e_t, msk, cov, a_out, nc_out);
}